// B2GravNetBlock_21887153340469
// MI455X (gfx1250) — compile-verified
//
#include <hip/hip_runtime.h>
#include <math.h>

// ---------------------------------------------------------------------------
// GravNet block for MI455X (gfx1250, wave32, WMMA).
// All GEMMs: V_WMMA_F32_16X16X4_F32, compile-time shapes, 4 row-tiles/wave.
// kNN stage: one workgroup per graph, S staged in LDS (64KB), broadcast loop.
// ---------------------------------------------------------------------------

typedef __attribute__((ext_vector_type(2))) float v2f;
typedef __attribute__((ext_vector_type(8))) float v8f;

#define NNODES 131072   // B * NPG
#define NPGC   1024
#define NGRAPH 128
#define SDIM   16
#define PDIM   64
#define KNN    5
#define BN_EPS 1e-5f

#define F_RELU 1
#define F_BN   2
#define F_ACC  4

// ---------------------------------------------------------------------------
// C[M,NCOLS] = epilogue(A[M,K] @ W[K,NCOLS] + bias)
// Block = 256 threads (8 waves). NCT = NCOLS/16 column tiles; the 8 waves
// split into 8/NCT row groups x NCT column tiles. Each wave computes RPW=4
// row tiles (64 rows) sharing one B fetch per k-step -> 4 WMMAs per B load.
// Rows per block: 512/NCT. Requires M % (512/NCT) == 0.
// Epilogue order: +bias -> (+C if ACC) -> relu -> eval-BN.
// ---------------------------------------------------------------------------
template <int K, int NCOLS, int FLAGS>
__global__ __launch_bounds__(256) void wmma_gemm(
    const float* __restrict__ A, const float* __restrict__ W,
    const float* __restrict__ bias,
    const float* __restrict__ bng, const float* __restrict__ bnb,
    const float* __restrict__ bnm, const float* __restrict__ bnv,
    float* __restrict__ C)
{
    constexpr int NCT  = NCOLS / 16;   // column tiles (1, 4, or 8)
    constexpr int RG   = 8 / NCT;      // row groups per block
    constexpr int RPW  = 4;            // row tiles per wave
    constexpr int ROWS = RG * RPW * 16;

    const int lane = threadIdx.x & 31;
    const int wave = threadIdx.x >> 5;
    const int ct   = wave % NCT;
    const int rg   = wave / NCT;
    const int r    = lane & 15;        // row-in-tile (A) / col-in-tile (B,C)
    const int hi   = lane >> 4;        // K-half (A,B) / M-half (C,D) selector

    const int n0 = ct * 16;
    const int m0 = blockIdx.x * ROWS + rg * (RPW * 16);

    // Per-lane base pointers; all k-step offsets become immediates.
    const float* ap[RPW];
    #pragma unroll
    for (int i = 0; i < RPW; ++i)
        ap[i] = A + (size_t)(m0 + i * 16 + r) * K + 2 * hi;
    const float* bp = W + (size_t)(2 * hi) * NCOLS + n0 + r;

    v8f c[RPW] = {};
    #pragma unroll
    for (int k = 0; k < K; k += 4) {
        v2f b;                                   // B 4x16: K pair by lane half
        b.x = bp[(size_t)k * NCOLS];
        b.y = bp[(size_t)k * NCOLS + NCOLS];
        v2f a[RPW];
        #pragma unroll
        for (int i = 0; i < RPW; ++i)            // A 16x4: contiguous K pair
            a[i] = *reinterpret_cast<const v2f*>(ap[i] + k);
        #pragma unroll
        for (int i = 0; i < RPW; ++i)            // 4 independent accumulators
            c[i] = __builtin_amdgcn_wmma_f32_16x16x4_f32(
                       false, a[i], false, b, (short)0, c[i], false, false);
    }

    const int n = n0 + r;
    const float bv = bias ? bias[n] : 0.0f;
    float scale = 1.0f, shift = 0.0f;
    if (FLAGS & F_BN) {
        const float sc = bng[n] * rsqrtf(bnv[n] + BN_EPS);
        scale = sc;
        shift = bnb[n] - bnm[n] * sc;
    }

    #pragma unroll
    for (int i = 0; i < RPW; ++i) {
        #pragma unroll
        for (int e = 0; e < 8; ++e) {            // C layout: VGPR e -> M=e(+8 hi)
            const int row = m0 + i * 16 + e + 8 * hi;
            float y = c[i][e] + bv;
            float* p = C + (size_t)row * NCOLS + n;
            if (FLAGS & F_ACC)  y += *p;
            if (FLAGS & F_RELU) y = fmaxf(y, 0.0f);
            y = y * scale + shift;
            *p = y;
        }
    }
}

// ---------------------------------------------------------------------------
// Per-graph kNN (k=5, includes self) in learned space coords.
// One workgroup (1024 threads, 32 waves) per graph. S block (1024x16 f32,
// 64KB) staged in LDS; inner-loop reads are wave-wide broadcasts.
// ---------------------------------------------------------------------------
__global__ __launch_bounds__(NPGC) void knn_topk(
    const float* __restrict__ S, int* __restrict__ nidx,
    float* __restrict__ nwgt)
{
    __shared__ float sh[NPGC * SDIM];           // 64 KB of 320 KB WGP LDS
    const int g   = blockIdx.x;
    const int tid = threadIdx.x;
    const float* Sg = S + (size_t)g * NPGC * SDIM;

    for (int i = tid; i < NPGC * SDIM / 4; i += NPGC)
        reinterpret_cast<float4*>(sh)[i] =
            reinterpret_cast<const float4*>(Sg)[i];
    __syncthreads();

    float si[SDIM];
    #pragma unroll
    for (int d = 0; d < SDIM; ++d) si[d] = sh[tid * SDIM + d];

    float bestd[KNN];
    int   besti[KNN];
    #pragma unroll
    for (int k = 0; k < KNN; ++k) { bestd[k] = 3.4e38f; besti[k] = 0; }

    for (int j = 0; j < NPGC; ++j) {
        float d2 = 0.0f;
        #pragma unroll
        for (int d = 0; d < SDIM; ++d) {
            const float t = si[d] - sh[j * SDIM + d];   // broadcast read
            d2 = fmaf(t, t, d2);
        }
        if (d2 < bestd[KNN - 1]) {
            int p = KNN - 1;
            #pragma unroll
            for (int q = KNN - 1; q > 0; --q) {
                if (bestd[q - 1] > d2) {
                    bestd[q] = bestd[q - 1];
                    besti[q] = besti[q - 1];
                    p = q - 1;
                }
            }
            bestd[p] = d2;
            besti[p] = j;
        }
    }

    const int node = g * NPGC + tid;
    #pragma unroll
    for (int k = 0; k < KNN; ++k) {
        nidx[node * KNN + k] = g * NPGC + besti[k];
        nwgt[node * KNN + k] = __expf(-10.0f * bestd[k]);
    }
}

// ---------------------------------------------------------------------------
// Weighted message aggregation: agg[node] = [mean_k(w_k*f_jk) | max_k(...)]
// One thread per (node, feature). Gathers stay in the graph's 256KB f-slab.
// ---------------------------------------------------------------------------
__global__ __launch_bounds__(256) void gather_agg(
    const float* __restrict__ F, const int* __restrict__ nidx,
    const float* __restrict__ nwgt, float* __restrict__ agg, int N)
{
    const int t = blockIdx.x * blockDim.x + threadIdx.x;
    if (t >= N * PDIM) return;
    const int node = t >> 6;
    const int p    = t & (PDIM - 1);

    float sum = 0.0f, mx = -3.4e38f;
    #pragma unroll
    for (int k = 0; k < KNN; ++k) {
        const int   j = nidx[node * KNN + k];
        const float w = nwgt[node * KNN + k];
        const float v = w * F[(size_t)j * PDIM + p];
        sum += v;
        mx = fmaxf(mx, v);
    }
    agg[(size_t)node * 128 + p]        = sum * (1.0f / KNN);
    agg[(size_t)node * 128 + PDIM + p] = mx;
}

// ---------------------------------------------------------------------------
extern "C" void kernel_launch(void* const* d_in, const int* in_sizes, int n_in,
                              void* d_out, int out_size, void* d_ws, size_t ws_size,
                              hipStream_t stream)
{
    const float* x   = (const float*)d_in[0];
    // d_in[1] = batch (implicit: contiguous equal-size graphs)
    const float* W1  = (const float*)d_in[2];
    const float* b1  = (const float*)d_in[3];
    const float* g1  = (const float*)d_in[4];
    const float* be1 = (const float*)d_in[5];
    const float* m1  = (const float*)d_in[6];
    const float* v1  = (const float*)d_in[7];
    const float* W2  = (const float*)d_in[8];
    const float* b2  = (const float*)d_in[9];
    const float* g2  = (const float*)d_in[10];
    const float* be2 = (const float*)d_in[11];
    const float* m2  = (const float*)d_in[12];
    const float* v2  = (const float*)d_in[13];
    const float* W3  = (const float*)d_in[14];
    const float* b3  = (const float*)d_in[15];
    const float* Ws  = (const float*)d_in[16];
    const float* bs  = (const float*)d_in[17];
    const float* Wh  = (const float*)d_in[18];
    const float* bh  = (const float*)d_in[19];
    const float* Wo1 = (const float*)d_in[20];
    const float* Wo2 = (const float*)d_in[21];
    const float* bo2 = (const float*)d_in[22];
    const float* g3  = (const float*)d_in[23];
    const float* be3 = (const float*)d_in[24];
    const float* m3  = (const float*)d_in[25];
    const float* v3  = (const float*)d_in[26];

    const int N = NNODES;
    float* out = (float*)d_out;

    // Workspace layout (floats). Total = 3 * N * 128 * 4B = 192 MB.
    float* ws  = (float*)d_ws;
    float* h   = ws;                          // [N,128] final pre-gravnet h
    float* h1  = ws + (size_t)N * 128;        // [N,128]; later reused as agg
    float* h2  = ws + (size_t)2 * N * 128;    // [N,128]; later reused:
    float* f   = h2;                          //   [N,64]
    float* s   = h2 + (size_t)N * 64;         //   [N,16]
    int*   nix = (int*)(h2 + (size_t)N * 80); //   [N,5] int
    float* nw  = h2 + (size_t)N * 85;         //   [N,5]

    const dim3 blk(256);
    // rows per block = 512/NCT -> grid = N*NCT/512
    const dim3 g128(N / 64);    // NCOLS=128: 64 rows/block
    const dim3 g64 (N / 128);   // NCOLS=64: 128 rows/block
    const dim3 g16 (N / 512);   // NCOLS=16: 512 rows/block

    // pre-gravnet MLP
    wmma_gemm<16, 128, F_RELU | F_BN><<<g128, blk, 0, stream>>>(
        x,  W1, b1, g1, be1, m1, v1, h1);
    wmma_gemm<128, 128, F_RELU | F_BN><<<g128, blk, 0, stream>>>(
        h1, W2, b2, g2, be2, m2, v2, h2);
    wmma_gemm<128, 128, 0><<<g128, blk, 0, stream>>>(
        h2, W3, b3, nullptr, nullptr, nullptr, nullptr, h);

    // GravNet projections
    wmma_gemm<128, 16, 0><<<g16, blk, 0, stream>>>(
        h, Ws, bs, nullptr, nullptr, nullptr, nullptr, s);
    wmma_gemm<128, 64, 0><<<g64, blk, 0, stream>>>(
        h, Wh, bh, nullptr, nullptr, nullptr, nullptr, f);

    // kNN + weights, then aggregation (h1's buffer becomes agg)
    knn_topk<<<NGRAPH, NPGC, 0, stream>>>(s, nix, nw);
    gather_agg<<<(N * PDIM + 255) / 256, blk, 0, stream>>>(f, nix, nw, h1, N);

    // out = BN3( h @ Wo1 + agg @ Wo2 + bo2 )
    wmma_gemm<128, 128, 0><<<g128, blk, 0, stream>>>(
        h, Wo1, nullptr, nullptr, nullptr, nullptr, nullptr, out);
    wmma_gemm<128, 128, F_BN | F_ACC><<<g128, blk, 0, stream>>>(
        h1, Wo2, bo2, g3, be3, m3, v3, out);
}